// ConceptFlow_86320252715113
// MI455X (gfx1250) — compile-verified
//
#include <hip/hip_runtime.h>

#define FEAT    128
#define NNODES  50000
#define NEDGES  625000
#define NEG_SLOPE 0.2f

typedef __attribute__((ext_vector_type(2))) float v2f;
typedef __attribute__((ext_vector_type(8))) float v8f;

// ---------- helpers ----------
__device__ __forceinline__ float lrelu(float x) { return x > 0.f ? x : NEG_SLOPE * x; }

// order-preserving float<->int map so atomicMax(int) == float max
__device__ __forceinline__ int   f2ord(float f) { int i = __float_as_int(f); return i >= 0 ? i : (i ^ 0x7fffffff); }
__device__ __forceinline__ float ord2f(int i)   { return __int_as_float(i >= 0 ? i : (i ^ 0x7fffffff)); }

__device__ __forceinline__ void atomAddF(float* p, float v) {
  // lowers to global_atomic_add_f32 (no CAS loop) on gfx1250
  unsafeAtomicAdd(p, v);
}

// ---------- 1) h = act(in) @ W via V_WMMA_F32_16X16X4_F32 ----------
// block = 256 threads = 8 waves; block computes rows [16*bx, 16*bx+16) x all 128 cols.
// wave w owns the 16x16 C tile at columns [16w, 16w+16).
__global__ __launch_bounds__(256) void gat_gemm_wmma(
    const float* __restrict__ in, const float* __restrict__ W,
    float* __restrict__ h, int relu_in)
{
  __shared__ __align__(16) float As[16][132];   // 132 stride: conflict-free A-frag reads
  const int tid  = threadIdx.x;
  const int row0 = blockIdx.x * 16;

  // stage 16x128 A tile into LDS (coalesced), applying ReLU for layers 2/3
  for (int i = tid; i < 16 * FEAT; i += 256) {
    const int r = i >> 7, k = i & 127;
    float v = in[(row0 + r) * FEAT + k];
    if (relu_in) v = v > 0.f ? v : 0.f;
    As[r][k] = v;
  }
  __syncthreads();

  const int lane = tid & 31;
  const int wave = tid >> 5;
  const int hh   = lane >> 4;      // half-wave: 0 or 1
  const int l16  = lane & 15;
  const int n0   = wave * 16;

  const float* Wp = W + n0 + l16;  // column n0+l16, rows indexed by k*FEAT
  v8f c = {};

#pragma unroll
  for (int k0 = 0; k0 < FEAT; k0 += 4) {
    const int ka = k0 + hh * 2;    // lanes 0-15 -> K=k0,k0+1 ; lanes 16-31 -> K=k0+2,k0+3
    v2f a = *(const v2f*)&As[l16][ka];             // A[M=l16][ka, ka+1]
    v2f b;
    b.x = Wp[ka * FEAT];                           // B[K=ka  ][N=n0+l16]
    b.y = Wp[(ka + 1) * FEAT];                     // B[K=ka+1][N=n0+l16]
    c = __builtin_amdgcn_wmma_f32_16x16x4_f32(false, a, false, b, (short)0, c, false, false);
  }

  // C layout: VGPR r -> row (r + 8*hh), col n0+l16
  float* hp = h + (row0 + hh * 8) * FEAT + n0 + l16;
#pragma unroll
  for (int r = 0; r < 8; ++r) hp[r * FEAT] = c[r];
}

// ---------- 2) el = h@al, er = h@ar (one wave per node) ----------
__global__ __launch_bounds__(256) void gat_elr(
    const float* __restrict__ h, const float* __restrict__ al,
    const float* __restrict__ ar, float* __restrict__ el, float* __restrict__ er)
{
  const int node = blockIdx.x * 8 + (threadIdx.x >> 5);
  const int lane = threadIdx.x & 31;
  const float4 hv = ((const float4*)(h + node * FEAT))[lane];
  const float4 av = ((const float4*)al)[lane];
  const float4 rv = ((const float4*)ar)[lane];
  float sl = hv.x * av.x + hv.y * av.y + hv.z * av.z + hv.w * av.w;
  float sr = hv.x * rv.x + hv.y * rv.y + hv.z * rv.z + hv.w * rv.w;
#pragma unroll
  for (int m = 16; m >= 1; m >>= 1) {
    sl += __shfl_xor(sl, m, 32);
    sr += __shfl_xor(sr, m, 32);
  }
  if (lane == 0) { el[node] = sl; er[node] = sr; }
}

// ---------- per-layer state init ----------
__global__ void gat_init(float* __restrict__ acc, int* __restrict__ mkey,
                         float* __restrict__ denom)
{
  const int i = blockIdx.x * 256 + threadIdx.x;
  if (i < NNODES * FEAT) acc[i] = 0.f;
  if (i < NNODES) { mkey[i] = (int)0x80000000; denom[i] = 0.f; }
}

// ---------- 3a) segment max over dst ----------
__global__ void gat_edge_max(const int* __restrict__ src, const int* __restrict__ dst,
                             const float* __restrict__ el, const float* __restrict__ er,
                             int* __restrict__ mkey)
{
  const int eid = blockIdx.x * 256 + threadIdx.x;
  if (eid >= NEDGES) return;
  const int s = src[eid], d = dst[eid];
  atomicMax(mkey + d, f2ord(lrelu(el[s] + er[d])));
}

// ---------- 3b) segment sum of exp(e - m) over dst ----------
__global__ void gat_edge_expsum(const int* __restrict__ src, const int* __restrict__ dst,
                                const float* __restrict__ el, const float* __restrict__ er,
                                const int* __restrict__ mkey, float* __restrict__ denom)
{
  const int eid = blockIdx.x * 256 + threadIdx.x;
  if (eid >= NEDGES) return;
  const int s = src[eid], d = dst[eid];
  const float v = lrelu(el[s] + er[d]);
  atomAddF(denom + d, __expf(v - ord2f(mkey[d])));
}

// ---------- 4) acc[dst] += alpha * h[src]  (one wave per edge) ----------
__global__ __launch_bounds__(256) void gat_edge_aggregate(
    const int* __restrict__ src, const int* __restrict__ dst,
    const float* __restrict__ el, const float* __restrict__ er,
    const int* __restrict__ mkey, const float* __restrict__ denom,
    const float* __restrict__ h, float* __restrict__ acc)
{
  const int eid  = blockIdx.x * 8 + (threadIdx.x >> 5);
  const int lane = threadIdx.x & 31;
  const int s = src[eid], d = dst[eid];
  const float v     = lrelu(el[s] + er[d]);
  const float alpha = __expf(v - ord2f(mkey[d])) / denom[d];
  const float4 hv = ((const float4*)(h + s * FEAT))[lane];
  float* ap = acc + d * FEAT + lane * 4;
  atomAddF(ap + 0, alpha * hv.x);
  atomAddF(ap + 1, alpha * hv.y);
  atomAddF(ap + 2, alpha * hv.z);
  atomAddF(ap + 3, alpha * hv.w);
}

// ---------- 5) out = acc + b ----------
__global__ void gat_finalize(const float* __restrict__ acc, const float* __restrict__ b,
                             float* __restrict__ out)
{
  const int i = blockIdx.x * 256 + threadIdx.x;
  if (i >= NNODES * FEAT) return;
  out[i] = acc[i] + b[i & 127];
}

// ---------- driver ----------
static void run_layer(const float* in, int relu_in,
                      const float* W, const float* al, const float* ar, const float* b,
                      const int* src, const int* dst,
                      float* h, float* acc, int* mkey, float* denom,
                      float* el, float* er, float* out, hipStream_t stream)
{
  gat_gemm_wmma    <<<NNODES / 16,            256, 0, stream>>>(in, W, h, relu_in);
  gat_elr          <<<NNODES / 8,             256, 0, stream>>>(h, al, ar, el, er);
  gat_init         <<<(NNODES * FEAT) / 256,  256, 0, stream>>>(acc, mkey, denom);
  gat_edge_max     <<<(NEDGES + 255) / 256,   256, 0, stream>>>(src, dst, el, er, mkey);
  gat_edge_expsum  <<<(NEDGES + 255) / 256,   256, 0, stream>>>(src, dst, el, er, mkey, denom);
  gat_edge_aggregate<<<NEDGES / 8,            256, 0, stream>>>(src, dst, el, er, mkey, denom, h, acc);
  gat_finalize     <<<(NNODES * FEAT) / 256,  256, 0, stream>>>(acc, b, out);
}

extern "C" void kernel_launch(void* const* d_in, const int* in_sizes, int n_in,
                              void* d_out, int out_size, void* d_ws, size_t ws_size,
                              hipStream_t stream)
{
  const float* x   = (const float*)d_in[0];
  const int*   src = (const int*)  d_in[1];
  const int*   dst = (const int*)  d_in[2];
  const float* W1  = (const float*)d_in[3];
  const float* al1 = (const float*)d_in[4];
  const float* ar1 = (const float*)d_in[5];
  const float* b1  = (const float*)d_in[6];
  const float* W2  = (const float*)d_in[7];
  const float* al2 = (const float*)d_in[8];
  const float* ar2 = (const float*)d_in[9];
  const float* b2  = (const float*)d_in[10];
  const float* W3  = (const float*)d_in[11];
  const float* al3 = (const float*)d_in[12];
  const float* ar3 = (const float*)d_in[13];
  const float* b3  = (const float*)d_in[14];
  float* out = (float*)d_out;

  // workspace carve-up (~77.6 MB)
  char* p = (char*)d_ws;
  float* h     = (float*)p; p += (size_t)NNODES * FEAT * sizeof(float);
  float* acc   = (float*)p; p += (size_t)NNODES * FEAT * sizeof(float);
  float* y     = (float*)p; p += (size_t)NNODES * FEAT * sizeof(float);
  float* el    = (float*)p; p += (size_t)NNODES * sizeof(float);
  float* er    = (float*)p; p += (size_t)NNODES * sizeof(float);
  float* denom = (float*)p; p += (size_t)NNODES * sizeof(float);
  int*   mkey  = (int*)  p; p += (size_t)NNODES * sizeof(int);

  // layer 1: input x (no relu), output -> y
  run_layer(x, 0, W1, al1, ar1, b1, src, dst, h, acc, mkey, denom, el, er, y, stream);
  // layer 2: input relu(y), output -> y (y fully consumed by gemm before overwrite)
  run_layer(y, 1, W2, al2, ar2, b2, src, dst, h, acc, mkey, denom, el, er, y, stream);
  // layer 3: input relu(y), output -> d_out (no final relu)
  run_layer(y, 1, W3, al3, ar3, b3, src, dst, h, acc, mkey, denom, el, er, out, stream);
}